// ClippedRelativeSelfAttention_19894288515138
// MI455X (gfx1250) — compile-verified
//
#include <hip/hip_runtime.h>
#include <math.h>

// ---------------------------------------------------------------------------
// ClippedRelativeSelfAttention for MI455X (gfx1250), fp32 WMMA 16x16x4 path.
// B=8, L=101, DIM=1024, HEADS=8, DH=128, MFEAT=621, window 7x7 (49 taps).
// Workspace usage: ~11.3 MB (floats):
//   q_ws     (B,H,112,128)  = 917504
//   v_ws     (B,H,112,128)  = 917504
//   vsum     (B,H,128)      = 8192
//   G        (624,16)       = 9984   cols: 0..4 = rf[r]-mde (col4==0), 5 = mde
//   qd_ws    (B,H,112,8)    = 57344  cols: 0..4 = qdot[r], 5 = q_md
//   attn     (808,1024)     = 827392
//   proj_pad (624,128)      = 79872  zero-padded rows >= 621
// ---------------------------------------------------------------------------

#define NL 101
#define LP 112
#define DH 128
#define NH 8
#define NB 8
#define MFEAT 621
#define NTOK (NB * NL)   // 808

typedef __attribute__((ext_vector_type(2))) float v2f;
typedef __attribute__((ext_vector_type(8))) float v8f;

__device__ __forceinline__ v8f wmma4(v2f a, v2f b, v8f c) {
  // V_WMMA_F32_16X16X4_F32 : D(16x16 f32) = A(16x4 f32) * B(4x16 f32) + C
  return __builtin_amdgcn_wmma_f32_16x16x4_f32(
      false, a, false, b, (short)0, c, false, false);
}

__device__ __forceinline__ v8f v8zero() {
  v8f z = {0.f, 0.f, 0.f, 0.f, 0.f, 0.f, 0.f, 0.f};
  return z;
}

// ---------------------------------------------------------------------------
// Kernel 0: zero a float4 range (q_ws+v_ws pad rows must be 0 every launch)
// ---------------------------------------------------------------------------
__global__ void zero_kernel(float4* __restrict__ p, int n4) {
  int i = blockIdx.x * blockDim.x + threadIdx.x;
  if (i < n4) p[i] = make_float4(0.f, 0.f, 0.f, 0.f);
}

// ---------------------------------------------------------------------------
// Kernel 0b: proj_pad[j][d] = (j < 621) ? proj[j][d] : 0   (624 x 128)
// ---------------------------------------------------------------------------
__global__ __launch_bounds__(256) void projpad_kernel(
    const float* __restrict__ proj, float* __restrict__ proj_pad) {
  int e = blockIdx.x * blockDim.x + threadIdx.x;  // float4 slot, 624*32 total
  if (e >= 624 * 32) return;
  int j = e >> 5;
  int c = (e & 31) * 4;
  float4 val = make_float4(0.f, 0.f, 0.f, 0.f);
  if (j < MFEAT) val = *(const float4*)(proj + (size_t)j * DH + c);
  *(float4*)(proj_pad + (size_t)j * DH + c) = val;
}

// ---------------------------------------------------------------------------
// Kernel 1/5: tiled WMMA GEMM  C[M,N] = A[M,K] * W[N,K]^T + bias
// MODE==1: N=2048 (Wq rows 0..1023, Wv rows 1024..2047); outputs scattered
//          into (B,H,LP,DH) head layout for q_ws / v_ws.
// MODE==0: N=1024, plain row-major output (final projection into d_out).
// Block: 256 threads = 8 waves, tile 64(M) x 128(N), K staged in 16-chunks.
// ---------------------------------------------------------------------------
#define TM 64
#define TN 128
#define KC 16

template <int MODE>
__global__ __launch_bounds__(256) void gemm_kernel(
    const float* __restrict__ X,   // M x K row-major
    const float* __restrict__ W0,  // first weight block (N0 x K)
    const float* __restrict__ W1,  // second weight block (MODE==1)
    const float* __restrict__ b0, const float* __restrict__ b1,
    float* __restrict__ outA, float* __restrict__ outB,
    int M, int N, int K) {
  __shared__ __align__(16) float As[TM][KC + 2];
  __shared__ __align__(16) float Bs[TN][KC + 2];

  const int tid = threadIdx.x;
  const int lane = tid & 31;
  const int wave = tid >> 5;
  const int wm = (wave & 1) * 32;
  const int wn = (wave >> 1) * 32;
  const int ln = lane & 15;
  const int klo = (lane >> 4) * 2;
  const int rhalf = (lane >> 4) * 8;

  const int ntiles = N / TN;
  const int m0 = (blockIdx.x / ntiles) * TM;
  const int n0 = (blockIdx.x % ntiles) * TN;

  v8f acc[2][2];
  acc[0][0] = v8zero(); acc[0][1] = v8zero();
  acc[1][0] = v8zero(); acc[1][1] = v8zero();

  for (int k0 = 0; k0 < K; k0 += KC) {
    // stage A: 64x16 (256 float4 slots, one per thread)
    {
      int r = tid >> 2;
      int c = (tid & 3) * 4;
      float4 val = make_float4(0.f, 0.f, 0.f, 0.f);
      int gr = m0 + r;
      if (gr < M) val = *(const float4*)(X + (size_t)gr * K + k0 + c);
      As[r][c] = val.x; As[r][c + 1] = val.y;
      As[r][c + 2] = val.z; As[r][c + 3] = val.w;
    }
    // stage B: 128x16 (512 float4 slots, two per thread)
    for (int i = 0; i < 2; ++i) {
      int e = tid + i * 256;
      int r = e >> 2;
      int c = (e & 3) * 4;
      int gn = n0 + r;
      const float* src;
      if (MODE == 1)
        src = (gn < 1024) ? (W0 + (size_t)gn * K) : (W1 + (size_t)(gn - 1024) * K);
      else
        src = W0 + (size_t)gn * K;
      float4 val = *(const float4*)(src + k0 + c);
      Bs[r][c] = val.x; Bs[r][c + 1] = val.y;
      Bs[r][c + 2] = val.z; Bs[r][c + 3] = val.w;
    }
    __syncthreads();

    for (int ks = 0; ks < KC; ks += 4) {
      v2f a0 = *(const v2f*)&As[wm + ln][ks + klo];
      v2f a1 = *(const v2f*)&As[wm + 16 + ln][ks + klo];
      v2f bb0 = *(const v2f*)&Bs[wn + ln][ks + klo];
      v2f bb1 = *(const v2f*)&Bs[wn + 16 + ln][ks + klo];
      acc[0][0] = wmma4(a0, bb0, acc[0][0]);
      acc[0][1] = wmma4(a0, bb1, acc[0][1]);
      acc[1][0] = wmma4(a1, bb0, acc[1][0]);
      acc[1][1] = wmma4(a1, bb1, acc[1][1]);
    }
    __syncthreads();
  }

  // epilogue: bias + scatter
  for (int j = 0; j < 2; ++j) {
    int colg = n0 + wn + j * 16 + ln;
    float biasv;
    if (MODE == 1)
      biasv = (colg < 1024) ? b0[colg] : b1[colg - 1024];
    else
      biasv = b0[colg];
    for (int i = 0; i < 2; ++i) {
      for (int vr = 0; vr < 8; ++vr) {
        int gr = m0 + wm + i * 16 + vr + rhalf;
        if (gr < M) {
          float val = acc[i][j][vr] + biasv;
          if (MODE == 0) {
            outA[(size_t)gr * N + colg] = val;
          } else {
            int b = gr / NL, l = gr - b * NL;
            int jj = colg;
            float* dst = outA;
            if (jj >= 1024) { dst = outB; jj -= 1024; }
            int h = jj >> 7, d = jj & 127;
            dst[(((size_t)b * NH + h) * LP + l) * DH + d] = val;
          }
        }
      }
    }
  }
}

// ---------------------------------------------------------------------------
// Kernel 2: rf = softmax_kernel(rpe) (5x621, global max), build G (624x16):
//           G[m][r] = rf[r][m]-mde[m] (r<5, col4==0), G[m][5] = mde[m].
// ---------------------------------------------------------------------------
__global__ __launch_bounds__(256) void rf_g_kernel(
    const float* __restrict__ rpe, const float* __restrict__ proj,
    float* __restrict__ G) {
  __shared__ float dd[5][624];
  __shared__ float diag[5];
  __shared__ float red[256];
  const int tid = threadIdx.x;
  const float dn = exp2f(-1.75f);          // 128^-0.25
  const float ratio = rsqrtf((float)MFEAT);

  for (int idx = tid; idx < 5 * MFEAT; idx += 256) {
    int r = idx / MFEAT, j = idx - r * MFEAT;
    float s = 0.f;
    for (int d = 0; d < DH; ++d) s += rpe[r * DH + d] * proj[j * DH + d];
    dd[r][j] = s * dn;
  }
  if (tid < 5) {
    float s = 0.f;
    for (int d = 0; d < DH; ++d) { float x = rpe[tid * DH + d]; s += x * x; }
    diag[tid] = 0.5f * dn * dn * s;
  }
  __syncthreads();

  float m = -3.4e38f;
  for (int idx = tid; idx < 5 * MFEAT; idx += 256) {
    int r = idx / MFEAT, j = idx - r * MFEAT;
    m = fmaxf(m, dd[r][j]);
  }
  red[tid] = m;
  __syncthreads();
  for (int s = 128; s > 0; s >>= 1) {
    if (tid < s) red[tid] = fmaxf(red[tid], red[tid + s]);
    __syncthreads();
  }
  const float gmax = red[0];

  for (int j = tid; j < 624; j += 256) {
    float rfv[5];
    for (int r = 0; r < 5; ++r)
      rfv[r] = (j < MFEAT)
                   ? ratio * (__expf(dd[r][j] - diag[r] - gmax) + 1e-4f)
                   : 0.f;
    float mde = rfv[4];
    float g[16];
    for (int c = 0; c < 16; ++c) g[c] = 0.f;
    for (int r = 0; r < 5; ++r) g[r] = rfv[r] - mde;  // g[4] == 0 exactly
    g[5] = mde;
    for (int c = 0; c < 16; ++c) G[(size_t)j * 16 + c] = g[c];
  }
}

// ---------------------------------------------------------------------------
// Kernel 3: v_sum[b,h,d] = sum_l v[b,h,l,d]
// ---------------------------------------------------------------------------
__global__ __launch_bounds__(128) void vsum_kernel(
    const float* __restrict__ v_ws, float* __restrict__ vsum) {
  int bh = blockIdx.x, d = threadIdx.x;
  float s = 0.f;
  for (int l = 0; l < NL; ++l) s += v_ws[((size_t)bh * LP + l) * DH + d];
  vsum[(size_t)bh * DH + d] = s;
}

// ---------------------------------------------------------------------------
// Kernel 4: per (b,h): qf = softmax_kernel(q) fused with
//   qd[l][0..4] = sum_m qf[l][m]*(rf[r][m]-mde[m]),  qd[l][5] = sum qf*mde.
// Pass 1: WMMA dd tiles -> lane-local running max (single cross-lane
//         reduction AFTER the tile loop: 32 bpermutes/wave total).
// Pass 2: WMMA dd recompute, exp -> QF tile in LDS -> second WMMA vs G.
// Block: 256 thr = 8 waves; wave w owns L-rows [16w,16w+16) (wave 7 = pad).
// ---------------------------------------------------------------------------
__global__ __launch_bounds__(256) void qf_attn_kernel(
    const float* __restrict__ q_ws, const float* __restrict__ proj_pad,
    const float* __restrict__ G, float* __restrict__ qd_ws) {
  __shared__ __align__(16) float qs[128][130];
  __shared__ __align__(16) float qft[8][16][20];
  __shared__ float diag[128];

  const int bh = blockIdx.x;
  const int tid = threadIdx.x;
  const int lane = tid & 31;
  const int wave = tid >> 5;
  const int ln = lane & 15;
  const int klo = (lane >> 4) * 2;
  const int rhalf = (lane >> 4) * 8;
  const float dn = exp2f(-1.75f);
  const float ratio = rsqrtf((float)MFEAT);

  // stage q (scaled by dn); rows >= LP zero-padded
  for (int it = 0; it < 16; ++it) {
    int e = tid + it * 256;            // 4096 float4 slots = 128x32
    int r = e >> 5;
    int c = (e & 31) * 4;
    float4 val = make_float4(0.f, 0.f, 0.f, 0.f);
    if (r < LP) val = *(const float4*)(q_ws + ((size_t)bh * LP + r) * DH + c);
    qs[r][c] = val.x * dn; qs[r][c + 1] = val.y * dn;
    qs[r][c + 2] = val.z * dn; qs[r][c + 3] = val.w * dn;
  }
  __syncthreads();
  if (tid < 128) {
    float s = 0.f;
    for (int k = 0; k < DH; ++k) { float x = qs[tid][k]; s += x * x; }
    diag[tid] = 0.5f * s;              // dn already folded into qs
  }
  __syncthreads();

  const int mrow = wave * 16;
  const int NT = (MFEAT + 15) / 16;    // 39
  const bool jvalid = true;            // proj_pad rows >= 621 are zero

  // ---- pass 1: lane-local running max, one reduction at the end ----
  float lmax[8];
#pragma unroll
  for (int i = 0; i < 8; ++i) lmax[i] = -3.4e38f;

  for (int nt = 0; nt < NT; ++nt) {
    const int jb = nt * 16;
    v8f acc = v8zero();
    for (int k0 = 0; k0 < DH; k0 += 4) {
      v2f a = *(const v2f*)&qs[mrow + ln][k0 + klo];
      v2f b = *(const v2f*)(proj_pad + (size_t)(jb + ln) * DH + k0 + klo);
      acc = wmma4(a, b, acc);
    }
    if (jb + ln < MFEAT) {             // only last tile partially masks
#pragma unroll
      for (int i = 0; i < 8; ++i) lmax[i] = fmaxf(lmax[i], acc[i]);
    }
  }
  // single cross-lane max per row register (16-lane groups)
  float vmax[8];
#pragma unroll
  for (int i = 0; i < 8; ++i) {
    float v = lmax[i];
    v = fmaxf(v, __shfl_xor(v, 1, 16));
    v = fmaxf(v, __shfl_xor(v, 2, 16));
    v = fmaxf(v, __shfl_xor(v, 4, 16));
    v = fmaxf(v, __shfl_xor(v, 8, 16));
    vmax[i] = v;
  }

  // ---- pass 2: recompute, exp, and reduce against G via WMMA ----
  v8f acc2 = v8zero();
  for (int nt = 0; nt < NT; ++nt) {
    const int jb = nt * 16;
    const int j = jb + ln;
    v8f acc = v8zero();
    for (int k0 = 0; k0 < DH; k0 += 4) {
      v2f a = *(const v2f*)&qs[mrow + ln][k0 + klo];
      v2f b = *(const v2f*)(proj_pad + (size_t)j * DH + k0 + klo);
      acc = wmma4(a, b, acc);
    }
    // exp and stage QF tile
#pragma unroll
    for (int i = 0; i < 8; ++i) {
      int row = i + rhalf;
      int grow = mrow + row;
      float e = 0.f;
      if (j < MFEAT && grow < NL)
        e = ratio * (__expf(acc[i] - diag[grow] - vmax[i]) + 1e-4f);
      qft[wave][row][ln] = e;
    }
    __syncthreads();
    // acc2 += QF(16x16) * G_chunk(16x16): k-dim is the feature index
    for (int ks = 0; ks < 16; ks += 4) {
      v2f a = *(const v2f*)&qft[wave][ln][ks + klo];
      int mg = jb + ks + klo;
      v2f b;
      b.x = G[(size_t)mg * 16 + ln];
      b.y = G[(size_t)(mg + 1) * 16 + ln];
      acc2 = wmma4(a, b, acc2);
    }
    __syncthreads();
  }
  (void)jvalid;

  // store qdot/q_md: cols 0..5 (rest are exact zeros)
#pragma unroll
  for (int i = 0; i < 8; ++i) {
    int l = mrow + i + rhalf;
    if (l < NL && ln < 8)
      qd_ws[((size_t)bh * LP + l) * 8 + ln] = acc2[i];
  }
}

// ---------------------------------------------------------------------------
// Kernel 6: windowed gather + combine:
//   out[b,h,l,d] = D_inv * ( q_md * v_sum[d] + sum_k w_k * v[b,h,pos_k,d] )
//   D_inv = 1 / ( sum_k qdot[diffs_k] + L * q_md )   (qdot[4]==0 exactly)
// Geometry recomputed inline (DIFFS / OFFSETS of the reference).
// ---------------------------------------------------------------------------
__global__ __launch_bounds__(128) void gather_kernel(
    const float* __restrict__ v_ws, const float* __restrict__ vsum,
    const float* __restrict__ qd_ws, float* __restrict__ attn) {
  const int blk = blockIdx.x;
  const int bh = blk / NL;
  const int l = blk - bh * NL;
  const int d = threadIdx.x;

  __shared__ float qd[6];
  if (threadIdx.x < 6) qd[threadIdx.x] = qd_ws[((size_t)bh * LP + l) * 8 + threadIdx.x];
  __syncthreads();

  const int x_off = (l < 100) ? (l % 10) : 0;
  const int y_off = (l < 100) ? (l / 10) : 10;

  float acc = 0.f;
  float den = 0.f;
  for (int k = 0; k < 49; ++k) {
    int dx = (k % 7) - 3, dy = (k / 7) - 3;
    int xp = x_off + dx, yp = y_off + dy;
    int adx = dx < 0 ? -dx : dx;
    int ady = dy < 0 ? -dy : dy;
    int diff = adx + ady;
    bool valid = (xp >= 0) && (xp < 10) && (yp >= 0) && (yp < 10) && (diff < 4);
    int ds = valid ? diff : 4;
    float w = qd[ds];                  // qd[4] == 0 kills invalid taps
    den += w;
    int pos = l + dx + 10 * dy;        // OFFSETS[k] = dx + 10*dy
    if (pos >= 0 && pos < NL)
      acc += w * v_ws[((size_t)bh * LP + pos) * DH + d];
  }
  den += (float)NL * qd[5];
  float dinv = 1.f / den;
  float out = (qd[5] * vsum[(size_t)bh * DH + d] + acc) * dinv;

  int b = bh >> 3, h = bh & 7;
  attn[((size_t)(b * NL + l)) * 1024 + h * DH + d] = out;
}

// ---------------------------------------------------------------------------
extern "C" void kernel_launch(void* const* d_in, const int* in_sizes, int n_in,
                              void* d_out, int out_size, void* d_ws,
                              size_t ws_size, hipStream_t stream) {
  (void)in_sizes; (void)n_in; (void)out_size; (void)ws_size;
  const float* x    = (const float*)d_in[0];
  const float* Wq   = (const float*)d_in[1];
  const float* bq   = (const float*)d_in[2];
  const float* Wv   = (const float*)d_in[3];
  const float* bv   = (const float*)d_in[4];
  const float* rpe  = (const float*)d_in[5];
  const float* proj = (const float*)d_in[6];
  const float* Wo   = (const float*)d_in[7];
  const float* bo   = (const float*)d_in[8];
  float* out = (float*)d_out;

  float* ws = (float*)d_ws;
  float* q_ws     = ws;                       // 917504
  float* v_ws     = ws + 917504;              // 917504
  float* vsum     = ws + 1835008;             // 8192
  float* G        = ws + 1843200;             // 9984
  float* qd_ws    = ws + 1853184;             // 57344
  float* attn     = ws + 1910528;             // 827392
  float* proj_pad = ws + 2737920;             // 79872  (total 2817792 fl ~11.3MB)

  // 0) zero q_ws + v_ws (pad rows must be zero each launch); pad proj
  zero_kernel<<<1792, 256, 0, stream>>>((float4*)q_ws, (2 * 917504) / 4);
  projpad_kernel<<<78, 256, 0, stream>>>(proj, proj_pad);

  // 1) fused Q/V projection: (808x1024) x (2048x1024)^T, head-scattered
  gemm_kernel<1><<<13 * (2048 / TN), 256, 0, stream>>>(
      x, Wq, Wv, bq, bv, q_ws, v_ws, NTOK, 2048, 1024);

  // 2) rf / G
  rf_g_kernel<<<1, 256, 0, stream>>>(rpe, proj, G);

  // 3) v row sums
  vsum_kernel<<<NB * NH, 128, 0, stream>>>(v_ws, vsum);

  // 4) fused softmax-kernel features + qdot/q_md reductions
  qf_attn_kernel<<<NB * NH, 256, 0, stream>>>(q_ws, proj_pad, G, qd_ws);

  // 5) windowed gather + normalize -> attn (token-major)
  gather_kernel<<<NB * NH * NL, 128, 0, stream>>>(v_ws, vsum, qd_ws, attn);

  // 6) output projection: (808x1024) x (1024x1024)^T + bo -> d_out
  gemm_kernel<0><<<13 * (1024 / TN), 256, 0, stream>>>(
      attn, Wo, nullptr, bo, nullptr, out, nullptr, NTOK, 1024, 1024);
}